// MaskedRoIAlign_19172734009518
// MI455X (gfx1250) — compile-verified
//
#include <hip/hip_runtime.h>

// MaskedRoIAlign for MI455X (gfx1250, wave32).
// Separable-bilinear formulation -> two fp32 WMMA GEMM stages per box:
//   out[c,ph,pw] = sum_j WY[ph,j] * sum_i WX[pw,i] * (feat*mask)[c, row(j), col(i)]
// using V_WMMA_F32_16X16X4_F32 (M=16 pad of 7 bins, N=16 channels, K=28 in 7 steps).

typedef __attribute__((ext_vector_type(2))) float v2f;
typedef __attribute__((ext_vector_type(8))) float v8f;

#define NBATCH 2
#define CCH    256
#define HH     64
#define WWID   64
#define KBOX   64
#define PHB    7
#define PWB    7
#define SCALEF 0.25f
#define SRR    2
#define NSAMP  (PHB * SRR)    // 14 sample rows/cols
#define NSLOT  (NSAMP * 2)    // 28 tap slots per axis
#define WAVES  4              // waves per block, each owns a 16-channel tile

__global__ __launch_bounds__(32 * WAVES)
void MaskedRoIAlign_kernel(const float* __restrict__ feat,   // [N,C,H,W]
                           const float* __restrict__ boxes,  // [K,5]
                           const float* __restrict__ mask,   // [K,C,H,W]
                           float* __restrict__ out)          // [K,C,7,7]
{
    __shared__ int   s_yrow[NSLOT];
    __shared__ float s_ywt [NSLOT];
    __shared__ int   s_xcol[NSLOT];
    __shared__ float s_xwt [NSLOT];
    __shared__ float s_V[WAVES][PWB * NSLOT * 16];   // stage-1 result per wave

    const int k     = blockIdx.y;                     // box
    const int wave  = threadIdx.x >> 5;               // 0..WAVES-1
    const int lane  = threadIdx.x & 31;               // 0..31
    const int cbase = (blockIdx.x * WAVES + wave) * 16;

    // ---- box geometry (uniform) ----
    const int   bi = (int)boxes[k * 5 + 0];
    const float x1 = boxes[k * 5 + 1] * SCALEF;
    const float y1 = boxes[k * 5 + 2] * SCALEF;
    const float x2 = boxes[k * 5 + 3] * SCALEF;
    const float y2 = boxes[k * 5 + 4] * SCALEF;
    const float bin_w = fmaxf(x2 - x1, 1.0f) / PWB;
    const float bin_h = fmaxf(y2 - y1, 1.0f) / PHB;

    // ---- slot tables: tap row/col indices + folded weights (0.5 * w * valid) ----
    if (threadIdx.x < NSLOT) {
        const int   t    = threadIdx.x;
        const int   s    = t >> 1;              // sample index 0..13
        const int   tap  = t & 1;               // lo/hi tap
        const int   bin  = s >> 1;              // output bin 0..6
        const int   sub  = s & 1;               // sub-sample 0..1
        const float off  = ((float)sub + 0.5f) / (float)SRR;
        // Y axis
        {
            const float v  = y1 + ((float)bin + off) * bin_h;
            const bool  ok = (v >= -1.0f) && (v <= (float)HH);
            const float vc = fminf(fmaxf(v, 0.0f), (float)(HH - 1));
            const int   lo = (int)vc;
            const int   hi = (lo + 1 < HH) ? lo + 1 : HH - 1;
            const float fr = vc - (float)lo;
            s_yrow[t] = tap ? hi : lo;
            s_ywt [t] = (tap ? fr : 1.0f - fr) * (ok ? 0.5f : 0.0f);
        }
        // X axis
        {
            const float v  = x1 + ((float)bin + off) * bin_w;
            const bool  ok = (v >= -1.0f) && (v <= (float)WWID);
            const float vc = fminf(fmaxf(v, 0.0f), (float)(WWID - 1));
            const int   lo = (int)vc;
            const int   hi = (lo + 1 < WWID) ? lo + 1 : WWID - 1;
            const float fr = vc - (float)lo;
            s_xcol[t] = tap ? hi : lo;
            s_xwt [t] = (tap ? fr : 1.0f - fr) * (ok ? 0.5f : 0.0f);
        }
    }
    __syncthreads();

    const int c  = cbase + (lane & 15);    // channel handled by this lane (N index)
    const int kh = lane >> 4;              // K-half within a K=4 WMMA step
    const int m  = lane & 15;              // M index for A fragments

    // ---- A fragments for both stages, built ONCE (pw/j-invariant).
    //      For j0 = 4t + 2*kh, both j0>>2 and (j0+1)>>2 equal t, so the
    //      row-select predicate collapses to (m == t). Load the weights
    //      unconditionally (LDS broadcast) and select by multiply — avoids
    //      exec-masked loads in the hot loops. ----
    v2f afragX[7], afragY[7];
    int xc0[7], xc1[7];
#pragma unroll
    for (int t = 0; t < 7; ++t) {
        const int   i0  = 4 * t + 2 * kh;
        const float sel = (m == t) ? 1.0f : 0.0f;
        const float wx0 = s_xwt[i0];
        const float wx1 = s_xwt[i0 + 1];
        const float wy0 = s_ywt[i0];
        const float wy1 = s_ywt[i0 + 1];
        afragX[t][0] = sel * wx0;
        afragX[t][1] = sel * wx1;
        afragY[t][0] = sel * wy0;
        afragY[t][1] = sel * wy1;
        xc0[t] = s_xcol[i0];
        xc1[t] = s_xcol[i0 + 1];
    }

    const float* fch = feat + ((size_t)bi * CCH + c) * (HH * WWID);
    const float* mch = mask + ((size_t)k  * CCH + c) * (HH * WWID);

    // ---- stage 1: V[pw, j, c] = sum_i WX[pw,i] * (feat*mask)[c, row(j), col(i)] ----
    for (int j = 0; j < NSLOT; ++j) {
        const int    y  = s_yrow[j];
        const float* frow = fch + y * WWID;
        const float* mrow = mch + y * WWID;
        v8f acc = {};
#pragma unroll
        for (int t = 0; t < 7; ++t) {
            v2f b;
            b[0] = frow[xc0[t]] * mrow[xc0[t]];
            b[1] = frow[xc1[t]] * mrow[xc1[t]];
            acc = __builtin_amdgcn_wmma_f32_16x16x4_f32(
                false, afragX[t], false, b, (short)0, acc, false, false);
        }
        // D layout: lanes 0-15 hold M=r in VGPR r -> bins pw=0..6 live here
        if (lane < 16) {
#pragma unroll
            for (int r = 0; r < PWB; ++r)
                s_V[wave][(r * NSLOT + j) * 16 + m] = acc[r];
        }
    }
    __syncthreads();   // uniform across block

    // ---- stage 2: out[ph, c] = sum_j WY[ph,j] * V[pw, j, c] for each pw ----
    for (int pw = 0; pw < PWB; ++pw) {
        v8f acc = {};
#pragma unroll
        for (int t = 0; t < 7; ++t) {
            const int j0 = 4 * t + 2 * kh;
            v2f b;
            b[0] = s_V[wave][(pw * NSLOT + j0    ) * 16 + (lane & 15)];
            b[1] = s_V[wave][(pw * NSLOT + j0 + 1) * 16 + (lane & 15)];
            acc = __builtin_amdgcn_wmma_f32_16x16x4_f32(
                false, afragY[t], false, b, (short)0, acc, false, false);
        }
        if (lane < 16) {
#pragma unroll
            for (int r = 0; r < PHB; ++r)
                out[(((size_t)k * CCH + c) * PHB + r) * PWB + pw] = acc[r];
        }
    }
}

extern "C" void kernel_launch(void* const* d_in, const int* in_sizes, int n_in,
                              void* d_out, int out_size, void* d_ws, size_t ws_size,
                              hipStream_t stream) {
    const float* feat  = (const float*)d_in[0];   // [2,256,64,64]
    const float* boxes = (const float*)d_in[1];   // [64,5]
    const float* mask  = (const float*)d_in[2];   // [64,256,64,64]
    float*       out   = (float*)d_out;           // [64,256,7,7]

    dim3 grid((CCH / 16) / WAVES, KBOX);          // (4 channel-tile groups, 64 boxes)
    dim3 block(32 * WAVES);                       // 4 wave32 per block
    MaskedRoIAlign_kernel<<<grid, block, 0, stream>>>(feat, boxes, mask, out);
}